// VectorQuantEMA_71829033059130
// MI455X (gfx1250) — compile-verified
//
#include <hip/hip_runtime.h>
#include <hip/hip_bf16.h>
#include <cstdint>

// ---------------------------------------------------------------------------
// VQ-VAE EMA vector quantization for MI455X (gfx1250, wave32, WMMA).
//
// N=16384 z-rows (D=256) vs K=8192 codes: 69 GFLOP fp32 matmul + argmin.
// Exact fp32 tensor path: V_WMMA_F32_16X16X4_F32 (argmin is precision
// sensitive; no bf16 downconvert). Codebook chunks are double-buffered in
// LDS via GLOBAL_LOAD_ASYNC_TO_LDS_B128 (ASYNCcnt) so L2->LDS traffic hides
// behind the WMMA chain.
//
// argmin_k ||z-c_k||^2  ==  argmax_k ( 2*z.c_k - ||c_k||^2 )
// ---------------------------------------------------------------------------

typedef float v2f __attribute__((ext_vector_type(2)));
typedef float v8f __attribute__((ext_vector_type(8)));

// Pointer types for the async-to-LDS builtin: clang reports param 0 as
// "int __vector(4) __device__*" (= v4i addrspace(1)*), so match exactly.
typedef int v4i_t __attribute__((vector_size(16)));
typedef __attribute__((address_space(1))) v4i_t* g_v4i;
typedef __attribute__((address_space(3))) v4i_t* l_v4i;

#define N_ROWS   16384
#define DIM      256
#define K_CODES  8192
#define MT       64          // z rows per block
#define NT       64          // codebook rows per chunk
#define LDS_STRIDE 260       // 256 + 4 pad -> conflict-free ds reads
#define N_CHUNKS (K_CODES / NT)
#define F4_PER_THR 16        // NT*DIM floats / 256 threads / 4

// Async global->LDS b128 copy (CDNA5 path, ASYNCcnt-tracked).
__device__ __forceinline__ void async_b128(const float* gsrc, float* ldst) {
#if __has_builtin(__builtin_amdgcn_global_load_async_to_lds_b128)
    __builtin_amdgcn_global_load_async_to_lds_b128(
        (g_v4i)gsrc, (l_v4i)ldst, 0, 0);
#else
    asm volatile("global_load_async_to_lds_b128 %0, %1, off"
                 :: "v"((unsigned)(uintptr_t)ldst), "v"(gsrc) : "memory");
#endif
}

__device__ __forceinline__ void wait_async_le16() {
#if __has_builtin(__builtin_amdgcn_s_wait_asynccnt)
    __builtin_amdgcn_s_wait_asynccnt(16);
#else
    asm volatile("s_wait_asynccnt 16" ::: "memory");
#endif
}
__device__ __forceinline__ void wait_async_0() {
#if __has_builtin(__builtin_amdgcn_s_wait_asynccnt)
    __builtin_amdgcn_s_wait_asynccnt(0);
#else
    asm volatile("s_wait_asynccnt 0" ::: "memory");
#endif
}

// -------------------------------------------------------------- ||c||^2 ----
__global__ __launch_bounds__(256) void vq_cnorm_kernel(
    const float* __restrict__ cb, float* __restrict__ cnorms)
{
    const int wid  = blockIdx.x * 8 + (threadIdx.x >> 5);   // one wave / code
    const int lane = threadIdx.x & 31;
    const float* row = cb + (size_t)wid * DIM;
    float s = 0.0f;
#pragma unroll
    for (int i = 0; i < 2; ++i) {
        const int d = lane * 4 + i * 128;
        float4 v = *(const float4*)(row + d);
        s += v.x * v.x + v.y * v.y + v.z * v.z + v.w * v.w;
    }
#pragma unroll
    for (int m = 16; m >= 1; m >>= 1) s += __shfl_xor(s, m, 32);
    if (lane == 0) cnorms[wid] = s;
}

// ------------------------------------------------- WMMA argmin (fp32) ------
__global__ __launch_bounds__(256) void vq_argmin_kernel(
    const float* __restrict__ z, const float* __restrict__ cb,
    const float* __restrict__ cnorms,
    int* __restrict__ codes_i32, float* __restrict__ codes_f32)
{
    __shared__ float zt[MT * LDS_STRIDE];        // 64 x 256 z tile (padded)
    __shared__ float ct[2][NT * LDS_STRIDE];     // double-buffered cb chunk

    const int t    = threadIdx.x;
    const int wid  = t >> 5;        // 8 waves
    const int lane = t & 31;
    const int l16  = lane & 15;
    const int hi   = lane >> 4;     // K-half select per WMMA fp32 A/B layout
    const int rowBase = blockIdx.x * MT;

    // Kick off async load of chunk 0 before staging z (overlap).
#pragma unroll
    for (int i = 0; i < F4_PER_THR; ++i) {
        const int f = (t * F4_PER_THR + i) * 4;
        const int r = f >> 8, c = f & 255;
        async_b128(cb + (size_t)r * DIM + c, &ct[0][r * LDS_STRIDE + c]);
    }

    // Stage the z tile (64 rows) once: 16 float4 per thread.
#pragma unroll
    for (int i = 0; i < 16; ++i) {
        const int f = (t * 16 + i) * 4;          // flat float index
        const int r = f >> 8, c = f & 255;
        float4 v = *(const float4*)(z + (size_t)(rowBase + r) * DIM + c);
        *(float4*)(zt + r * LDS_STRIDE + c) = v;
    }

    const int mg = wid >> 1;        // 4 M-groups of 16 rows
    const int ng = wid & 1;         // 2 N-groups of 32 codes per chunk
    const int mlocal = mg * 16;

    // Running best (score, index) per lane: VGPR j of the 16x16 D-fragment
    // holds M = mbase + (hi?8:0) + j at column N = nbase + l16.
    float bs[8]; int bi[8];
#pragma unroll
    for (int j = 0; j < 8; ++j) { bs[j] = -3.4e38f; bi[j] = 0; }

    const float* za = zt + (mlocal + l16) * LDS_STRIDE + 2 * hi;

    for (int ch = 0; ch < N_CHUNKS; ++ch) {
        const int buf = ch & 1;
        // Prefetch chunk ch+1 into the other buffer (ASYNCcnt-tracked).
        if (ch + 1 < N_CHUNKS) {
            const int kNext = (ch + 1) * NT;
#pragma unroll
            for (int i = 0; i < F4_PER_THR; ++i) {
                const int f = (t * F4_PER_THR + i) * 4;
                const int r = f >> 8, c = f & 255;
                async_b128(cb + (size_t)(kNext + r) * DIM + c,
                           &ct[buf ^ 1][r * LDS_STRIDE + c]);
            }
            wait_async_le16();   // oldest 16 (this chunk) complete in order
        } else {
            wait_async_0();
        }
        __syncthreads();         // all waves' async data visible in LDS

        const int kBase = ch * NT;
        v8f acc0 = {}; v8f acc1 = {};
        const float* cb0 = ct[buf] + (ng * 32 + l16) * LDS_STRIDE + 2 * hi;
        const float* cb1 = cb0 + 16 * LDS_STRIDE;

#pragma unroll 8
        for (int d0 = 0; d0 < DIM; d0 += 4) {
            // fp32 WMMA A 16x4 layout: lanes<16 -> K=d0,d0+1; lanes>=16 -> K=d0+2,d0+3
            v2f a  = { za[d0],  za[d0 + 1]  };
            v2f b0 = { cb0[d0], cb0[d0 + 1] };
            v2f b1 = { cb1[d0], cb1[d0 + 1] };
            acc0 = __builtin_amdgcn_wmma_f32_16x16x4_f32(
                false, a, false, b0, (short)0, acc0, false, false);
            acc1 = __builtin_amdgcn_wmma_f32_16x16x4_f32(
                false, a, false, b1, (short)0, acc1, false, false);
        }

        const int n0 = kBase + ng * 32 + l16;
        const int n1 = n0 + 16;
        const float cn0 = cnorms[n0];
        const float cn1 = cnorms[n1];
#pragma unroll
        for (int j = 0; j < 8; ++j) {
            const float s0 = 2.0f * acc0[j] - cn0;   // maximize == argmin dist
            const float s1 = 2.0f * acc1[j] - cn1;
            if (s0 > bs[j]) { bs[j] = s0; bi[j] = n0; }   // strict > keeps
            if (s1 > bs[j]) { bs[j] = s1; bi[j] = n1; }   // first occurrence
        }
        __syncthreads();  // readers done before ch+2 overwrites ct[buf]
    }

    // Reduce across the 16 N-columns of each half-wave (masks stay in-half).
#pragma unroll
    for (int j = 0; j < 8; ++j) {
#pragma unroll
        for (int m = 8; m >= 1; m >>= 1) {
            const float os = __shfl_xor(bs[j], m, 32);
            const int   oi = __shfl_xor(bi[j], m, 32);
            if (os > bs[j] || (os == bs[j] && oi < bi[j])) { bs[j] = os; bi[j] = oi; }
        }
    }

    // Combine the two N-groups (waves 2g and 2g+1) through LDS.
    float* sScore = zt;               // reuse tiles
    int*   sIdx   = (int*)ct;
    __syncthreads();
    if (l16 == 0) {
#pragma unroll
        for (int j = 0; j < 8; ++j) {
            const int m = mlocal + hi * 8 + j;
            sScore[m * 2 + ng] = bs[j];
            sIdx[m * 2 + ng]   = bi[j];
        }
    }
    __syncthreads();
    if (t < MT) {
        const float s0 = sScore[t * 2], s1 = sScore[t * 2 + 1];
        const int   i0 = sIdx[t * 2],   i1 = sIdx[t * 2 + 1];
        const int best = (s1 > s0 || (s1 == s0 && i1 < i0)) ? i1 : i0;
        const int r = rowBase + t;
        codes_i32[r] = best;
        codes_f32[r] = (float)best;
    }
}

// ---------------------------------------- gather z_q + loss + EMA stats ----
__global__ __launch_bounds__(256) void vq_gather_kernel(
    const float* __restrict__ z, const float* __restrict__ cb,
    const int* __restrict__ codes, float* __restrict__ zq_out,
    float* __restrict__ sums, float* __restrict__ counts,
    float* __restrict__ lossAcc)
{
    const int n    = blockIdx.x * 8 + (threadIdx.x >> 5);  // one wave / row
    const int lane = threadIdx.x & 31;
    const int code = codes[n];
    const float* crow = cb + (size_t)code * DIM;
    const float* zrow = z  + (size_t)n    * DIM;
    float*       orow = zq_out + (size_t)n * DIM;
    float*       srow = sums   + (size_t)code * DIM;
    float lsum = 0.0f;
#pragma unroll
    for (int i = 0; i < 2; ++i) {
        const int d = lane * 4 + i * 128;
        float4 c4 = *(const float4*)(crow + d);
        float4 z4 = *(const float4*)(zrow + d);
        *(float4*)(orow + d) = c4;                     // z_q_st value == z_q
        const float dx = c4.x - z4.x, dy = c4.y - z4.y;
        const float dz = c4.z - z4.z, dw = c4.w - z4.w;
        lsum += dx * dx + dy * dy + dz * dz + dw * dw;
        atomicAdd(srow + d + 0, z4.x);
        atomicAdd(srow + d + 1, z4.y);
        atomicAdd(srow + d + 2, z4.z);
        atomicAdd(srow + d + 3, z4.w);
    }
#pragma unroll
    for (int m = 16; m >= 1; m >>= 1) lsum += __shfl_xor(lsum, m, 32);
    if (lane == 0) {
        atomicAdd(lossAcc, lsum);
        atomicAdd(counts + code, 1.0f);
    }
}

// ----------------------------------------------- EMA count + total sum -----
__global__ __launch_bounds__(256) void vq_count_kernel(
    const float* __restrict__ ema_count, const float* __restrict__ counts,
    float* __restrict__ new_count_out, float* __restrict__ nTotal)
{
    const int k = blockIdx.x * 256 + threadIdx.x;
    const float nc = ema_count[k] * 0.99f + 0.01f * counts[k];
    new_count_out[k] = nc;
    float s = nc;
#pragma unroll
    for (int m = 16; m >= 1; m >>= 1) s += __shfl_xor(s, m, 32);
    __shared__ float wsum[8];
    if ((threadIdx.x & 31) == 0) wsum[threadIdx.x >> 5] = s;
    __syncthreads();
    if (threadIdx.x == 0) {
        float tot = 0.0f;
#pragma unroll
        for (int i = 0; i < 8; ++i) tot += wsum[i];
        atomicAdd(nTotal, tot);
    }
}

// ------------------------------- new_mean / new_codebook / loss output -----
__global__ __launch_bounds__(256) void vq_final_kernel(
    const float* __restrict__ ema_mean, const float* __restrict__ sums,
    const float* __restrict__ new_count, const float* __restrict__ nTotal,
    const float* __restrict__ lossAcc,
    float* __restrict__ new_mean_out, float* __restrict__ new_cb_out,
    float* __restrict__ loss_out)
{
    const int k = blockIdx.x;          // 8192
    const int d = threadIdx.x;         // 256
    const float ntot = nTotal[0];
    const float nc = new_count[k];
    const float cs = (nc + 1e-5f) / (ntot + (float)K_CODES * 1e-5f) * ntot;
    const size_t idx = (size_t)k * DIM + d;
    const float nm = ema_mean[idx] * 0.99f + 0.01f * sums[idx];
    new_mean_out[idx] = nm;
    new_cb_out[idx]   = nm / cs;
    if (k == 0 && d == 0)
        loss_out[0] = 1.25f * lossAcc[0] / ((float)N_ROWS * (float)DIM);
}

// ---------------------------------------------------------------------------
extern "C" void kernel_launch(void* const* d_in, const int* in_sizes, int n_in,
                              void* d_out, int out_size, void* d_ws, size_t ws_size,
                              hipStream_t stream) {
    const float* z         = (const float*)d_in[0];
    const float* cb        = (const float*)d_in[1];
    const float* ema_count = (const float*)d_in[2];
    const float* ema_mean  = (const float*)d_in[3];

    float* out = (float*)d_out;
    float* o_zq       = out;                       // 16384*256
    float* o_loss     = out + 4194304;             // 1
    float* o_codes    = out + 4194305;             // 16384 (float-cast ints)
    float* o_newcb    = out + 4210689;             // 8192*256
    float* o_newcount = out + 6307841;             // 8192
    float* o_newmean  = out + 6316033;             // 8192*256

    char* ws = (char*)d_ws;
    float* lossAcc = (float*)(ws + 0);
    float* nTotal  = (float*)(ws + 256);
    int*   codes_i = (int*)  (ws + 512);                         // 64 KB
    float* counts  = (float*)(ws + 512 + 65536);                 // 32 KB
    float* cnorms  = (float*)(ws + 512 + 65536 + 32768);         // 32 KB
    float* sums    = (float*)(ws + 512 + 65536 + 32768 + 32768); // 8 MB

    // Zero accumulators (loss, nTotal, counts, sums) every call: the harness
    // poisons scratch once and graph-replays without re-initializing.
    (void)hipMemsetAsync(d_ws, 0, 512 + 65536 + 32768 + 32768 + 8388608, stream);

    vq_cnorm_kernel <<<K_CODES / 8, 256, 0, stream>>>(cb, cnorms);
    vq_argmin_kernel<<<N_ROWS / MT, 256, 0, stream>>>(z, cb, cnorms, codes_i, o_codes);
    vq_gather_kernel<<<N_ROWS / 8, 256, 0, stream>>>(z, cb, codes_i, o_zq, sums, counts, lossAcc);
    vq_count_kernel <<<K_CODES / 256, 256, 0, stream>>>(ema_count, counts, o_newcount, nTotal);
    vq_final_kernel <<<K_CODES, 256, 0, stream>>>(ema_mean, sums, o_newcount, nTotal,
                                                  lossAcc, o_newmean, o_newcb, o_loss);
}